// GradientLoss_26242250179439
// MI455X (gfx1250) — compile-verified
//
#include <hip/hip_runtime.h>
#include <cstddef>

// ---------------- problem constants ----------------
#define BB 16
#define HH 768
#define WW 1024
#define TX 128            // output tile width
#define TY 16             // output tile height
#define LSTR 132          // LDS row stride (floats), 130 used + pad
#define LROWS 18          // TY + 2 halo rows
#define TILES_X 8         // ceil((WW-2)/TX)
#define TILES_Y 48        // ceil((HH-2)/TY)
#define DEPTH_VALID_F 0.001f
#define EPS_F 1e-8f

// ---------------- CDNA5 async global->LDS copy ----------------
// Probe feedback: builtins exist; param types are
//   b128: (AS1 int __attribute__((vector_size(16)))*, AS3 same, imm int, imm int)
//   b32 : (AS1 int*, AS3 int*, imm int, imm int)
typedef int v4i_t __attribute__((vector_size(4 * sizeof(int))));

typedef __attribute__((address_space(1))) v4i_t* g128_t;
typedef __attribute__((address_space(3))) v4i_t* l128_t;
typedef __attribute__((address_space(1))) int*   g32_t;
typedef __attribute__((address_space(3))) int*   l32_t;

__device__ __forceinline__ void async_copy_b128(const float* __restrict__ g, float* l) {
#if __has_builtin(__builtin_amdgcn_global_load_async_to_lds_b128)
  __builtin_amdgcn_global_load_async_to_lds_b128((g128_t)(g), (l128_t)(l), 0, 0);
#else
  asm volatile("global_load_async_to_lds_b128 %0, %1, off"
               :: "v"((unsigned)(unsigned long long)(l)),
                  "v"((unsigned long long)(g))
               : "memory");
#endif
}

__device__ __forceinline__ void async_copy_b32(const float* __restrict__ g, float* l) {
#if __has_builtin(__builtin_amdgcn_global_load_async_to_lds_b32)
  __builtin_amdgcn_global_load_async_to_lds_b32((g32_t)(g), (l32_t)(l), 0, 0);
#else
  asm volatile("global_load_async_to_lds_b32 %0, %1, off"
               :: "v"((unsigned)(unsigned long long)(l)),
                  "v"((unsigned long long)(g))
               : "memory");
#endif
}

__device__ __forceinline__ void async_wait_all() {
#if __has_builtin(__builtin_amdgcn_s_wait_asynccnt)
  __builtin_amdgcn_s_wait_asynccnt(0);
#else
  asm volatile("s_wait_asynccnt 0" ::: "memory");
#endif
}

// ---------------- kernel 1: fused stencil + per-block partial sums ----------------
__global__ __launch_bounds__(256) void grad_loss_tile(
    const float* __restrict__ sr, const float* __restrict__ hr,
    float* __restrict__ partials)
{
  __shared__ float s_sr[LROWS * LSTR];
  __shared__ float s_hr[LROWS * LSTR];
  __shared__ float s_red[3 * 256];

  const int tid = threadIdx.x;
  const int bz  = blockIdx.z;                      // batch
  const int x0  = 1 + blockIdx.x * TX;             // first output column
  const int y0  = 1 + blockIdx.y * TY;             // first output row
  const size_t base = (size_t)bz * (size_t)(HH * WW);

  // ---- DMA tile (rows y0-1 .. y0+TY, cols x0-1 .. x0+TX) into LDS ----
  // Body: cols [0,127] of the tile as 32 x 16B chunks per row (16B aligned
  // both in global (x0-1 is a multiple of 128) and in LDS (LSTR*4 = 528 = 33*16)).
  for (int e = tid; e < LROWS * 32; e += 256) {
    const int row = e >> 5;
    const int chk = e & 31;
    int gy = y0 - 1 + row; gy = gy > HH - 1 ? HH - 1 : gy;   // clamp (garbage only feeds predicated-off outputs)
    const int gx = (x0 - 1) + chk * 4;                       // always <= 1020, +3 stays in bounds
    const size_t gi = base + (size_t)gy * WW + gx;
    const int li = row * LSTR + chk * 4;
    async_copy_b128(sr + gi, s_sr + li);
    async_copy_b128(hr + gi, s_hr + li);
  }
  // Tail: tile cols 128,129 (right halo), 2 per row per array.
  if (tid < 2 * LROWS * 2) {
    const int arr = tid / (LROWS * 2);
    const int r2  = tid % (LROWS * 2);
    const int row = r2 >> 1;
    const int col = 128 + (r2 & 1);
    int gy = y0 - 1 + row; gy = gy > HH - 1 ? HH - 1 : gy;
    int gx = (x0 - 1) + col; gx = gx > WW - 1 ? WW - 1 : gx;
    const size_t gi = base + (size_t)gy * WW + gx;
    const int li = row * LSTR + col;
    if (arr == 0) async_copy_b32(sr + gi, s_sr + li);
    else          async_copy_b32(hr + gi, s_hr + li);
  }
  async_wait_all();      // each wave waits for its own async copies (ASYNCcnt)
  __syncthreads();       // cross-wave visibility of LDS tile

  // ---- compute: each thread walks 8 rows of one column with a rolling window ----
  const int c  = tid & (TX - 1);        // tile-local column (LDS cols c..c+2)
  const int r0 = (tid >> 7) * 8;        // 0 or 8: first of 8 output rows
  const int x  = x0 + c;

  float nv = 0.f, sx = 0.f, sy = 0.f;

  float h0a, h0b, h0c, h1a, h1b, h1c, h2a, h2b, h2c;
  float d0a, d0b, d0c, d1a, d1b, d1c, d2a, d2b, d2c;

  {
    int i = r0 * LSTR + c;
    h0a = s_hr[i]; h0b = s_hr[i + 1]; h0c = s_hr[i + 2];
    d0a = s_sr[i] - h0a; d0b = s_sr[i + 1] - h0b; d0c = s_sr[i + 2] - h0c;
    i += LSTR;
    h1a = s_hr[i]; h1b = s_hr[i + 1]; h1c = s_hr[i + 2];
    d1a = s_sr[i] - h1a; d1b = s_sr[i + 1] - h1b; d1c = s_sr[i + 2] - h1c;
  }

#pragma unroll
  for (int k = 0; k < 8; ++k) {
    const int i = (r0 + k + 2) * LSTR + c;
    h2a = s_hr[i]; h2b = s_hr[i + 1]; h2c = s_hr[i + 2];
    d2a = s_sr[i] - h2a; d2b = s_sr[i + 1] - h2b; d2c = s_sr[i + 2] - h2c;

    const int y = y0 + r0 + k;
    bool valid = (h0a > DEPTH_VALID_F) & (h0b > DEPTH_VALID_F) & (h0c > DEPTH_VALID_F) &
                 (h1a > DEPTH_VALID_F) & (h1b > DEPTH_VALID_F) & (h1c > DEPTH_VALID_F) &
                 (h2a > DEPTH_VALID_F) & (h2b > DEPTH_VALID_F) & (h2c > DEPTH_VALID_F);
    valid = valid && (y <= HH - 2) && (x <= WW - 2);

    // Sobel on diff = sr - hr (linearity); abs makes conv/corr flip irrelevant.
    const float gx = (d0a - d0c) + 2.f * (d1a - d1c) + (d2a - d2c);
    const float gy = (d0a + 2.f * d0b + d0c) - (d2a + 2.f * d2b + d2c);
    if (valid) {
      nv += 1.f;
      sx += fabsf(gx);
      sy += fabsf(gy);
    }

    // roll window down one row
    h0a = h1a; h0b = h1b; h0c = h1c;  d0a = d1a; d0b = d1b; d0c = d1c;
    h1a = h2a; h1b = h2b; h1c = h2c;  d1a = d2a; d1b = d2b; d1c = d2c;
  }

  // ---- deterministic block tree reduction ----
  s_red[tid]       = nv;
  s_red[256 + tid] = sx;
  s_red[512 + tid] = sy;
  __syncthreads();
#pragma unroll
  for (int s = 128; s > 0; s >>= 1) {
    if (tid < s) {
      s_red[tid]       += s_red[tid + s];
      s_red[256 + tid] += s_red[256 + tid + s];
      s_red[512 + tid] += s_red[512 + tid + s];
    }
    __syncthreads();
  }
  if (tid == 0) {
    const int tile = blockIdx.y * TILES_X + blockIdx.x;
    const int slot = (bz * (TILES_X * TILES_Y) + tile) * 3;
    partials[slot + 0] = s_red[0];
    partials[slot + 1] = s_red[256];
    partials[slot + 2] = s_red[512];
  }
}

// ---------------- kernel 2: deterministic final reduction -> scalar ----------------
__global__ __launch_bounds__(64) void grad_loss_finalize(
    const float* __restrict__ partials, float* __restrict__ out)
{
  __shared__ float s[BB];
  const int tid = threadIdx.x;
  const int tilesPerBatch = TILES_X * TILES_Y;     // 384
  if (tid < BB) {
    float nv = 0.f, sx = 0.f, sy = 0.f;
    for (int t = 0; t < tilesPerBatch; ++t) {      // fixed order => deterministic
      const int i = (tid * tilesPerBatch + t) * 3;
      nv += partials[i + 0];
      sx += partials[i + 1];
      sy += partials[i + 2];
    }
    s[tid] = (sx + sy) / (nv + EPS_F);             // loss_x_b + loss_y_b
  }
  __syncthreads();
  if (tid == 0) {
    float acc = 0.f;
    for (int b = 0; b < BB; ++b) acc += s[b];
    out[0] = acc * (1.0f / (float)BB);             // mean over batch
  }
}

// ---------------- host launcher ----------------
extern "C" void kernel_launch(void* const* d_in, const int* in_sizes, int n_in,
                              void* d_out, int out_size, void* d_ws, size_t ws_size,
                              hipStream_t stream) {
  (void)in_sizes; (void)n_in; (void)out_size; (void)ws_size;
  const float* sr = (const float*)d_in[0];
  const float* hr = (const float*)d_in[1];
  float* out      = (float*)d_out;
  float* partials = (float*)d_ws;                  // needs 16*384*3*4 = 73728 B

  dim3 grid(TILES_X, TILES_Y, BB);
  grad_loss_tile<<<grid, 256, 0, stream>>>(sr, hr, partials);
  grad_loss_finalize<<<1, 64, 0, stream>>>(partials, out);
}